// HypergraphConv_65481071395026
// MI455X (gfx1250) — compile-verified
//
#include <hip/hip_runtime.h>
#include <hip/hip_bf16.h>

typedef __attribute__((ext_vector_type(2))) float v2f;
typedef __attribute__((ext_vector_type(8))) float v8f;

#define EPSF 1e-9f

#if defined(__AMDGCN__) && __has_builtin(__builtin_amdgcn_global_load_async_to_lds_b128) && \
    __has_builtin(__builtin_amdgcn_s_wait_asynccnt)
#define HAS_ASYNC_LDS 1
typedef int v4i_ __attribute__((vector_size(4 * sizeof(int))));
typedef __attribute__((address_space(1))) v4i_* gbl_v4i_ptr;   // global src (AS1)
typedef __attribute__((address_space(3))) v4i_* lds_v4i_ptr;   // LDS dst (AS3)
#else
#define HAS_ASYNC_LDS 0
#endif

static __device__ __forceinline__ v8f wmma4(v2f a, v2f b, v8f c) {
    // V_WMMA_F32_16X16X4_F32 : D = A(16x4) x B(4x16) + C(16x16), full fp32
    return __builtin_amdgcn_wmma_f32_16x16x4_f32(
        /*neg_a=*/false, a, /*neg_b=*/false, b,
        /*c_mod=*/(short)0, c, /*reuse_a=*/false, /*reuse_b=*/false);
}

// ---------------------------------------------------------------------------
// Reductions: Dv_inv[b][n] = 1/(sum_e H + eps), De_inv[b][e] = 1/(sum_n H + eps)
// ---------------------------------------------------------------------------
__global__ __launch_bounds__(256) void row_sum_inv_kernel(
    const float* __restrict__ H, float* __restrict__ dvInv, int rows, int E) {
    int waveId = blockIdx.x * (blockDim.x >> 5) + (threadIdx.x >> 5);
    int lane = threadIdx.x & 31;
    if (waveId >= rows) return;
    const float* row = H + (size_t)waveId * E;
    float s = 0.0f;
    for (int c = lane * 4; c < E; c += 32 * 4) {
        float4 v = *(const float4*)(row + c);
        s += v.x + v.y + v.z + v.w;
    }
    #pragma unroll
    for (int off = 16; off > 0; off >>= 1) s += __shfl_xor(s, off, 32);
    if (lane == 0) dvInv[waveId] = 1.0f / (s + EPSF);
}

__global__ __launch_bounds__(256) void col_sum_inv_kernel(
    const float* __restrict__ H, float* __restrict__ deInv, int N, int E) {
    int e = blockIdx.x * blockDim.x + threadIdx.x;
    int b = blockIdx.y;
    const float* Hb = H + (size_t)b * N * E;
    float s = 0.0f;
    #pragma unroll 8
    for (int n = 0; n < N; ++n) s += Hb[(size_t)n * E + e];
    deInv[(size_t)b * E + e] = 1.0f / (s + EPSF);
}

// ---------------------------------------------------------------------------
// FP32 WMMA GEMM: C[b] = op(A[b]) @ B[b]  (+ per-row scale, + optional relu)
//   TRANS_A=0: A(m,k) = A[m*lda + k]  -> A tile staged with async global->LDS
//   TRANS_A=1: A(m,k) = A[k*lda + m]  -> H^T via LDS transpose (register path)
// Block = 128 threads (4 waves), 64x64 output tile, K in chunks of 32.
// Double-buffered LDS; B stored transposed (Bt[n][k]) so every fragment is an
// aligned ds_load_b64; per-k-step fragment prefetch pipelines LDS latency.
// ---------------------------------------------------------------------------
template <bool TRANS_A, bool RELU>
__global__ __launch_bounds__(128) void wmma_gemm_f32(
    const float* __restrict__ Ag, size_t strideA,
    const float* __restrict__ Bg, size_t strideB,
    float* __restrict__ Cg, size_t strideC,
    const float* __restrict__ scaleG, size_t strideScale,
    int K, int lda, int ldb, int ldc) {
    // pad: 36 -> 16B-aligned b128 LDS writes (direct/async path), conflict-free reads
    //      38 -> scatter-store path, 8B-aligned b64 reads, conflict-free reads
    constexpr int APAD = TRANS_A ? 38 : 36;
    constexpr int BPAD = 38;
    constexpr bool ASYNC_A = (!TRANS_A) && (HAS_ASYNC_LDS != 0);

    const int bz = blockIdx.z;
    const float* A = Ag + (size_t)bz * strideA;
    const float* Bp = Bg + (size_t)bz * strideB;
    float* C = Cg + (size_t)bz * strideC;
    const float* scale = scaleG ? (scaleG + (size_t)bz * strideScale) : nullptr;

    const int n0 = blockIdx.x * 64;
    const int m0 = blockIdx.y * 64;

    __shared__ float As[2][64][APAD];
    __shared__ float Bt[2][64][BPAD];   // transposed: Bt[n][k]

    const int t = threadIdx.x;
    const int wave = t >> 5;
    const int lane = t & 31;
    const int half = (lane >> 4) & 1;
    const int r = lane & 15;
    const int moff = (wave >> 1) * 32;
    const int noff = (wave & 1) * 32;

    // per-thread global-tile coordinates
    const int aRow = TRANS_A ? (t >> 4) : (t >> 3);          // trans: 0..7, else 0..15
    const int aC4  = TRANS_A ? ((t & 15) * 4) : ((t & 7) * 4);
    const int bRow = t >> 4;                                  // 0..7
    const int bC4  = (t & 15) * 4;

    float4 aReg[4], bReg[4];

    // async DMA of the A tile straight into (padded) LDS rows
    auto asyncLoadA = [&](int kk, int buf) {
#if HAS_ASYNC_LDS
        #pragma unroll
        for (int p = 0; p < 4; ++p) {
            const float* g = A + (size_t)(m0 + p * 16 + aRow) * lda + (kk + aC4);
            float* l = &As[buf][p * 16 + aRow][aC4];
            __builtin_amdgcn_global_load_async_to_lds_b128(
                (gbl_v4i_ptr)(uintptr_t)g,
                (lds_v4i_ptr)(uint32_t)(uintptr_t)l, 0, 0);
        }
#else
        (void)kk; (void)buf;
#endif
    };
    auto gloadA = [&](int kk) {
        #pragma unroll
        for (int p = 0; p < 4; ++p) {
            if (!TRANS_A)
                aReg[p] = *(const float4*)(A + (size_t)(m0 + p * 16 + aRow) * lda + (kk + aC4));
            else
                aReg[p] = *(const float4*)(A + (size_t)(kk + p * 8 + aRow) * lda + (m0 + aC4));
        }
    };
    auto gloadB = [&](int kk) {
        #pragma unroll
        for (int p = 0; p < 4; ++p)
            bReg[p] = *(const float4*)(Bp + (size_t)(kk + p * 8 + bRow) * ldb + (n0 + bC4));
    };
    auto gprefetch = [&](int kk) {   // global_prefetch_b8, two chunks ahead
        #pragma unroll
        for (int p = 0; p < 4; ++p) {
            if (!TRANS_A)
                __builtin_prefetch((const void*)(A + (size_t)(m0 + p * 16 + aRow) * lda + (kk + aC4)), 0, 3);
            else
                __builtin_prefetch((const void*)(A + (size_t)(kk + p * 8 + aRow) * lda + (m0 + aC4)), 0, 3);
            __builtin_prefetch((const void*)(Bp + (size_t)(kk + p * 8 + bRow) * ldb + (n0 + bC4)), 0, 3);
        }
    };
    auto storeA = [&](int buf) {
        #pragma unroll
        for (int p = 0; p < 4; ++p) {
            if (!TRANS_A) {
                *(float4*)(&As[buf][p * 16 + aRow][aC4]) = aReg[p];
            } else {
                const float* av = reinterpret_cast<const float*>(&aReg[p]);
                const int rr = p * 8 + aRow;
                As[buf][aC4 + 0][rr] = av[0];
                As[buf][aC4 + 1][rr] = av[1];
                As[buf][aC4 + 2][rr] = av[2];
                As[buf][aC4 + 3][rr] = av[3];
            }
        }
    };
    auto storeB = [&](int buf) {
        #pragma unroll
        for (int p = 0; p < 4; ++p) {
            const float* bv = reinterpret_cast<const float*>(&bReg[p]);
            const int rr = p * 8 + bRow;
            Bt[buf][bC4 + 0][rr] = bv[0];
            Bt[buf][bC4 + 1][rr] = bv[1];
            Bt[buf][bC4 + 2][rr] = bv[2];
            Bt[buf][bC4 + 3][rr] = bv[3];
        }
    };
    auto loadFrags = [&](int buf, int k2, v2f& a0, v2f& a1, v2f& b0, v2f& b1) {
        const int kb = k2 + 2 * half;   // fragment layout: K = 2*half + j
        a0 = *(const v2f*)(&As[buf][moff + r][kb]);
        a1 = *(const v2f*)(&As[buf][moff + 16 + r][kb]);
        b0 = *(const v2f*)(&Bt[buf][noff + r][kb]);
        b1 = *(const v2f*)(&Bt[buf][noff + 16 + r][kb]);
    };

    v8f acc[2][2] = {};
    const int nChunks = K >> 5;

    // prologue: stage chunk 0 into buffer 0
    if (ASYNC_A) asyncLoadA(0, 0); else { gloadA(0); }
    gloadB(0);
    if (!ASYNC_A) storeA(0);
    storeB(0);
#if HAS_ASYNC_LDS
    if (ASYNC_A) __builtin_amdgcn_s_wait_asynccnt(0);
#endif
    __syncthreads();

    for (int c = 0; c < nChunks; ++c) {
        const int cur = c & 1;
        const bool hasNext = (c + 1) < nChunks;
        if (hasNext) {
            // buffer cur^1 is free (barrier at end of previous iteration)
            if (ASYNC_A) asyncLoadA((c + 1) * 32, cur ^ 1); else gloadA((c + 1) * 32);
            gloadB((c + 1) * 32);
            if (c + 2 < nChunks) gprefetch((c + 2) * 32);
        }

        v2f a0[2], a1[2], b0[2], b1[2];
        loadFrags(cur, 0, a0[0], a1[0], b0[0], b1[0]);
        #pragma unroll
        for (int k2 = 0; k2 < 8; ++k2) {
            const int p = k2 & 1, q = p ^ 1;
            if (k2 < 7) loadFrags(cur, (k2 + 1) * 4, a0[q], a1[q], b0[q], b1[q]);
            acc[0][0] = wmma4(a0[p], b0[p], acc[0][0]);
            acc[0][1] = wmma4(a0[p], b1[p], acc[0][1]);
            acc[1][0] = wmma4(a1[p], b0[p], acc[1][0]);
            acc[1][1] = wmma4(a1[p], b1[p], acc[1][1]);
        }

        if (hasNext) {
            if (!ASYNC_A) storeA(cur ^ 1);
            storeB(cur ^ 1);
#if HAS_ASYNC_LDS
            if (ASYNC_A) __builtin_amdgcn_s_wait_asynccnt(0);
#endif
            __syncthreads();
        }
    }

    // epilogue: row scale + optional relu; C VGPR i -> row (i + 8*half)
    #pragma unroll
    for (int mi = 0; mi < 2; ++mi) {
        #pragma unroll
        for (int i = 0; i < 8; ++i) {
            const int gm = m0 + moff + mi * 16 + i + 8 * half;
            const float s = scale ? scale[gm] : 1.0f;
            #pragma unroll
            for (int ni = 0; ni < 2; ++ni) {
                float v = acc[mi][ni][i] * s;
                if (RELU) v = fmaxf(v, 0.0f);
                C[(size_t)gm * ldc + (n0 + noff + ni * 16 + r)] = v;
            }
        }
    }
}

// ---------------------------------------------------------------------------
extern "C" void kernel_launch(void* const* d_in, const int* in_sizes, int n_in,
                              void* d_out, int out_size, void* d_ws, size_t ws_size,
                              hipStream_t stream) {
    (void)in_sizes; (void)n_in; (void)out_size; (void)ws_size;
    const float* x = (const float*)d_in[0];  // [B, N, F]
    const float* H = (const float*)d_in[1];  // [B, N, E]
    const float* W = (const float*)d_in[2];  // [F, F]
    float* out = (float*)d_out;              // [B, N, F]

    const int B = 8, N = 2048, E = 1024, F = 256;

    float* XW    = (float*)d_ws;                   // B*N*F
    float* HTX   = XW + (size_t)B * N * F;         // B*E*F (De_inv-scaled)
    float* DvInv = HTX + (size_t)B * E * F;        // B*N
    float* DeInv = DvInv + (size_t)B * N;          // B*E

    // degree reductions
    row_sum_inv_kernel<<<dim3((B * N) / 8), 256, 0, stream>>>(H, DvInv, B * N, E);
    col_sum_inv_kernel<<<dim3(E / 256, B), 256, 0, stream>>>(H, DeInv, N, E);

    // XW = x @ W            (M=N, Nn=F, K=F); W shared across batch (strideB=0)
    wmma_gemm_f32<false, false><<<dim3(F / 64, N / 64, B), 128, 0, stream>>>(
        x, (size_t)N * F, W, (size_t)0, XW, (size_t)N * F,
        nullptr, 0, /*K=*/F, /*lda=*/F, /*ldb=*/F, /*ldc=*/F);

    // HTX = De_inv ⊙ (H^T @ XW)   (M=E, Nn=F, K=N); A transposed, lda = E
    wmma_gemm_f32<true, false><<<dim3(F / 64, E / 64, B), 128, 0, stream>>>(
        H, (size_t)N * E, XW, (size_t)N * F, HTX, (size_t)E * F,
        DeInv, (size_t)E, /*K=*/N, /*lda=*/E, /*ldb=*/F, /*ldc=*/F);

    // out = relu(Dv_inv ⊙ (H @ HTX))   (M=N, Nn=F, K=E)
    wmma_gemm_f32<false, true><<<dim3(F / 64, N / 64, B), 128, 0, stream>>>(
        H, (size_t)N * E, HTX, (size_t)E * F, out, (size_t)N * F,
        DvInv, (size_t)N, /*K=*/E, /*lda=*/E, /*ldb=*/F, /*ldc=*/F);
}